// Scattering_Third_Order_Real_2d_11166914970309
// MI455X (gfx1250) — compile-verified
//
#include <hip/hip_runtime.h>
#include <math.h>

// ---------------------------------------------------------------------------
// Third-order 2-D scattering, reduced via Parseval to:
//   per (b,f):  Xf = F_f * FFT2(x_b);  xa = |IFFT2(Xf)|_eps;  Xa = FFT2(xa)
//               G  = Xf.re*Xa.re + Xf.im*Xa.im            (real, 128x128)
//   out_full = (1/MN^2) * G(66x16384) @ filters^T(16384x33)   <-- fp32 WMMA
//   gather 672 (i,j) pairs per batch.
// ---------------------------------------------------------------------------

#define FFT_N   128
#define ROWS    16          // rows (or cols) of length-128 FFT per workgroup
#define TPB     256
#define B_      2
#define NF      33
#define NIMG    (B_ * NF)   // 66
#define MN      16384
#define CPITCH  129         // padded LDS pitch for column pass (bank-conflict free)
#define EPS_    1e-6f
#define PI_     3.14159265358979323846f

typedef float v2f __attribute__((ext_vector_type(2)));
typedef float v8f __attribute__((ext_vector_type(8)));

// ---- Stockham radix-2 DIF: 16 independent 128-point FFTs held in LDS. ----
// sgn = -1 forward, +1 inverse (unscaled). Returns buffer holding the result.
__device__ __forceinline__ float2* fft128_stages(float2* a, float2* b, int pitch, float sgn)
{
    float2* s = a;
    float2* d = b;
    int l = 64, lgm = 0;
    #pragma unroll
    for (int st = 0; st < 7; ++st) {
        const int m = 1 << lgm;
        __syncthreads();
        #pragma unroll 1
        for (int t = threadIdx.x; t < ROWS * 64; t += TPB) {
            const int row = t >> 6;          // which of the 16 transforms
            const int idx = t & 63;          // butterfly id; l*m == 64 always
            const int j = idx >> lgm;
            const int k = idx & (m - 1);
            const float2 c0 = s[row * pitch + k + m * j];
            const float2 c1 = s[row * pitch + k + m * j + 64];
            const float ang = sgn * (PI_ / (float)l) * (float)j;   // w_{2l}^j
            float wi, wr;
            __sincosf(ang, &wi, &wr);
            float2 sum, tw;
            sum.x = c0.x + c1.x; sum.y = c0.y + c1.y;
            const float dx = c0.x - c1.x, dy = c0.y - c1.y;
            tw.x = dx * wr - dy * wi;
            tw.y = dx * wi + dy * wr;
            d[row * pitch + k + 2 * m * j]     = sum;
            d[row * pitch + k + 2 * m * j + m] = tw;
        }
        float2* tmp = s; s = d; d = tmp;
        l >>= 1; ++lgm;
    }
    __syncthreads();
    return s;
}

// ---- Row passes (rows are contiguous in memory) ---------------------------

// real input -> forward row FFT (used on x)
__global__ void __launch_bounds__(TPB)
k_rows_real(const float* __restrict__ x, float2* __restrict__ X)
{
    __shared__ float2 A[ROWS * FFT_N];
    __shared__ float2 Bb[ROWS * FFT_N];
    const size_t base = (size_t)blockIdx.x * (ROWS * FFT_N);
    for (int t = threadIdx.x; t < ROWS * FFT_N; t += TPB) {
        float2 v; v.x = x[base + t]; v.y = 0.f;
        A[t] = v;
    }
    float2* r = fft128_stages(A, Bb, FFT_N, -1.f);
    for (int t = threadIdx.x; t < ROWS * FFT_N; t += TPB) X[base + t] = r[t];
}

// Xf = filters[f] * X[b]; store Xf; inverse row FFT -> T[b,f]
__global__ void __launch_bounds__(TPB)
k_rows_mod(const float2* __restrict__ X, const float* __restrict__ filt,
           float2* __restrict__ XF, float2* __restrict__ T)
{
    __shared__ float2 A[ROWS * FFT_N];
    __shared__ float2 Bb[ROWS * FFT_N];
    const int bf = blockIdx.x >> 3;
    const int rb = blockIdx.x & 7;
    const int b  = bf / NF;
    const int f  = bf - b * NF;
    const size_t off = (size_t)rb * (ROWS * FFT_N);
    const float2* src = X    + (size_t)b  * MN + off;
    const float*  fl  = filt + (size_t)f  * MN + off;
    float2*       xfo = XF   + (size_t)bf * MN + off;
    float2*       to  = T    + (size_t)bf * MN + off;
    for (int t = threadIdx.x; t < ROWS * FFT_N; t += TPB) {
        float2 v = src[t];
        const float w = fl[t];
        v.x *= w; v.y *= w;
        xfo[t] = v;
        A[t] = v;
    }
    float2* r = fft128_stages(A, Bb, FFT_N, +1.f);
    for (int t = threadIdx.x; t < ROWS * FFT_N; t += TPB) to[t] = r[t];
}

// in-place forward row FFT (used on xa)
__global__ void __launch_bounds__(TPB)
k_rows_fwd(float2* __restrict__ T)
{
    __shared__ float2 A[ROWS * FFT_N];
    __shared__ float2 Bb[ROWS * FFT_N];
    const size_t base = (size_t)blockIdx.x * (ROWS * FFT_N);
    for (int t = threadIdx.x; t < ROWS * FFT_N; t += TPB) A[t] = T[base + t];
    float2* r = fft128_stages(A, Bb, FFT_N, -1.f);
    for (int t = threadIdx.x; t < ROWS * FFT_N; t += TPB) T[base + t] = r[t];
}

// ---- Column passes (stride-128 in memory; transposed into padded LDS) ----

// in-place forward column FFT (finishes FFT2(x))
__global__ void __launch_bounds__(TPB)
k_cols_fwd(float2* __restrict__ X)
{
    __shared__ float2 A[ROWS * CPITCH];
    __shared__ float2 Bb[ROWS * CPITCH];
    const int img = blockIdx.x >> 3;
    const int cb  = blockIdx.x & 7;
    const size_t base = (size_t)img * MN + (size_t)cb * ROWS;
    for (int t = threadIdx.x; t < ROWS * FFT_N; t += TPB) {
        const int c = t & 15, r = t >> 4;
        A[c * CPITCH + r] = X[base + (size_t)r * FFT_N + c];
    }
    float2* res = fft128_stages(A, Bb, CPITCH, -1.f);
    for (int t = threadIdx.x; t < ROWS * FFT_N; t += TPB) {
        const int c = t & 15, r = t >> 4;
        X[base + (size_t)r * FFT_N + c] = res[c * CPITCH + r];
    }
}

// inverse column FFT, apply 1/MN, epilogue xa = sqrt(re^2+im^2+eps) -> (xa,0)
__global__ void __launch_bounds__(TPB)
k_cols_inv_abs(float2* __restrict__ T)
{
    __shared__ float2 A[ROWS * CPITCH];
    __shared__ float2 Bb[ROWS * CPITCH];
    const int img = blockIdx.x >> 3;
    const int cb  = blockIdx.x & 7;
    const size_t base = (size_t)img * MN + (size_t)cb * ROWS;
    for (int t = threadIdx.x; t < ROWS * FFT_N; t += TPB) {
        const int c = t & 15, r = t >> 4;
        A[c * CPITCH + r] = T[base + (size_t)r * FFT_N + c];
    }
    float2* res = fft128_stages(A, Bb, CPITCH, +1.f);
    const float s = 1.0f / (float)MN;
    for (int t = threadIdx.x; t < ROWS * FFT_N; t += TPB) {
        const int c = t & 15, r = t >> 4;
        const float2 v = res[c * CPITCH + r];
        const float re = v.x * s, im = v.y * s;
        float2 o; o.x = sqrtf(re * re + im * im + EPS_); o.y = 0.f;
        T[base + (size_t)r * FFT_N + c] = o;
    }
}

// forward column FFT (finishes FFT2(xa)); epilogue G = Re(conj(Xf)*Xa)
__global__ void __launch_bounds__(TPB)
k_cols_fwd_g(const float2* __restrict__ T, const float2* __restrict__ XF,
             float* __restrict__ G)
{
    __shared__ float2 A[ROWS * CPITCH];
    __shared__ float2 Bb[ROWS * CPITCH];
    const int img = blockIdx.x >> 3;
    const int cb  = blockIdx.x & 7;
    const size_t base = (size_t)img * MN + (size_t)cb * ROWS;
    for (int t = threadIdx.x; t < ROWS * FFT_N; t += TPB) {
        const int c = t & 15, r = t >> 4;
        A[c * CPITCH + r] = T[base + (size_t)r * FFT_N + c];
    }
    float2* res = fft128_stages(A, Bb, CPITCH, -1.f);
    for (int t = threadIdx.x; t < ROWS * FFT_N; t += TPB) {
        const int c = t & 15, r = t >> 4;
        const float2 v  = res[c * CPITCH + r];
        const float2 xf = XF[base + (size_t)r * FFT_N + c];
        G[base + (size_t)r * FFT_N + c] = xf.x * v.x + xf.y * v.y;
    }
}

// ---- fp32 WMMA GEMM: OUTF(80x48 padded) = (1/MN^2) * G(66xK) @ F^T(Kx33) ----
// 15 blocks (5 m-tiles x 3 n-tiles), 8 waves/block splitting K=16384,
// v_wmma_f32_16x16x4_f32 accumulation, LDS sum of partial tiles.
__global__ void __launch_bounds__(TPB)
k_gemm_wmma(const float* __restrict__ G, const float* __restrict__ F,
            float* __restrict__ OUTF)
{
    __shared__ float red[8 * 256];
    const int lane = threadIdx.x & 31;
    const int wave = threadIdx.x >> 5;
    const int mt = blockIdx.x / 3;
    const int nt = blockIdx.x - mt * 3;

    const int mrow = mt * 16 + (lane & 15);
    const int ncol = nt * 16 + (lane & 15);
    const int arow = (mrow < NIMG) ? mrow : 0;   // clamp padding rows
    const int brow = (ncol < NF)   ? ncol : 0;
    const int kk = (lane >> 4) << 1;             // K sub-offset per ISA layout

    const float* Ap = G + (size_t)arow * MN + kk;
    const float* Bp = F + (size_t)brow * MN + kk;

    v8f c = {0.f, 0.f, 0.f, 0.f, 0.f, 0.f, 0.f, 0.f};
    const int k0 = wave * (MN / 8);
    #pragma unroll 4
    for (int k = k0; k < k0 + MN / 8; k += 4) {
        const v2f a = *(const v2f*)(Ap + k);     // A[m][k..k+1]
        const v2f b = *(const v2f*)(Bp + k);     // B[k..k+1][n] = F[n][k..k+1]
        c = __builtin_amdgcn_wmma_f32_16x16x4_f32(
                false, a, false, b, (short)0, c, false, false);
    }

    #pragma unroll
    for (int r = 0; r < 8; ++r) red[wave * 256 + r * 32 + lane] = c[r];
    __syncthreads();

    const int e = threadIdx.x;                   // one tile element per thread
    float acc = 0.f;
    #pragma unroll
    for (int w = 0; w < 8; ++w) acc += red[w * 256 + e];

    const int r  = e >> 5;
    const int le = e & 31;
    const int ml = ((le >> 4) << 3) + r;         // C/D layout: M=(lane>>4)*8+r
    const int nl = le & 15;                      // N = lane&15
    const float scale = 1.0f / ((float)MN * (float)MN);
    OUTF[(size_t)(mt * 16 + ml) * 48 + (nt * 16 + nl)] = acc * scale;
}

// ---- gather the 672 (i,j) index pairs per batch ---------------------------
__global__ void __launch_bounds__(TPB)
k_gather(const float* __restrict__ OUTF, float* __restrict__ out)
{
    const int t = blockIdx.x * blockDim.x + threadIdx.x;
    if (t >= B_ * 672) return;
    const int b = t / 672;
    int rem = t - b * 672;
    int i = 0, jj = 0;
    bool found = false;
    for (int j1 = 0; j1 < 4 && !found; ++j1) {
        const int per = (4 - j1) * 8 + 1;        // entries per i at scale j1
        for (int l1 = 0; l1 < 8 && !found; ++l1) {
            if (rem < per) {
                i = j1 * 8 + l1;
                if (rem == per - 1) jj = NF - 1;                 // lowpass
                else jj = (j1 + (rem >> 3)) * 8 + (rem & 7);
                found = true;
            } else {
                rem -= per;
            }
        }
    }
    out[t] = OUTF[(size_t)(b * NF + i) * 48 + jj];
}

// ---------------------------------------------------------------------------
extern "C" void kernel_launch(void* const* d_in, const int* in_sizes, int n_in,
                              void* d_out, int out_size, void* d_ws, size_t ws_size,
                              hipStream_t stream)
{
    (void)in_sizes; (void)n_in; (void)out_size; (void)ws_size;
    const float* x       = (const float*)d_in[0];   // (2,1,128,128) f32
    const float* filters = (const float*)d_in[1];   // (1,33,128,128) f32
    float* out = (float*)d_out;                     // (2,672) f32

    // workspace layout (~22 MB)
    float2* X    = (float2*)d_ws;                   // 2   x 16384 complex
    float2* XF   = X  + (size_t)B_   * MN;          // 66  x 16384 complex
    float2* T    = XF + (size_t)NIMG * MN;          // 66  x 16384 complex
    float*  G    = (float*)(T + (size_t)NIMG * MN); // 66  x 16384 real
    float*  OUTF = G + (size_t)NIMG * MN;           // 80  x 48 real (padded)

    k_rows_real  <<<B_  * 8, TPB, 0, stream>>>(x, X);            // fft rows of x
    k_cols_fwd   <<<B_  * 8, TPB, 0, stream>>>(X);               // fft cols -> X
    k_rows_mod   <<<NIMG * 8, TPB, 0, stream>>>(X, filters, XF, T); // Xf + ifft rows
    k_cols_inv_abs<<<NIMG * 8, TPB, 0, stream>>>(T);             // ifft cols + |.|_eps
    k_rows_fwd   <<<NIMG * 8, TPB, 0, stream>>>(T);              // fft rows of xa
    k_cols_fwd_g <<<NIMG * 8, TPB, 0, stream>>>(T, XF, G);       // fft cols + G
    k_gemm_wmma  <<<15, TPB, 0, stream>>>(G, filters, OUTF);     // WMMA GEMM
    k_gather     <<<6,  TPB, 0, stream>>>(OUTF, out);            // (B,672)
}